// RegionEmbeddingLayer_48885317763663
// MI455X (gfx1250) — compile-verified
//
#include <hip/hip_runtime.h>
#include <hip/hip_bf16.h>
#include <stdint.h>

// Problem constants (from reference): B=16, S=2048, E=128, R=5, V=50000
#define B_ 16
#define S_ 2048
#define E_ 128
#define R_ 5

#define TS    32              // s-positions per block
#define ROWS  (TS + 4)        // staged emb rows (halo of 2 on each side)
#define WAVES 8
#define THREADS (WAVES * 32)
#define ITER  (TS / WAVES)    // s-positions per wave

typedef float v4f __attribute__((ext_vector_type(4)));

__global__ __launch_bounds__(THREADS)
void region_embed_kernel(const int* __restrict__ seq,
                         const float* __restrict__ seq_emb,
                         const float* __restrict__ U,
                         float* __restrict__ out)
{
    __shared__ float tile[ROWS * E_];   // 18 KB of the 320 KB/WGP LDS

    const int tid  = threadIdx.x;
    const int lane = tid & 31;          // wave32
    const int wave = tid >> 5;

    const int tiles_per_b = S_ / TS;
    const int b  = blockIdx.x / tiles_per_b;
    const int s0 = (blockIdx.x % tiles_per_b) * TS;

    // ---- Stage emb rows [s0-2, s0+TS+2) into LDS with async global->LDS copies.
    // One wave moves a whole 512 B row per instruction: 32 lanes x b128.
    // Out-of-range halo rows are zero-filled so the compute loop needs no
    // bounds checks at all (0 * u == reference's padded contribution).
    const float* emb_base = seq_emb + (size_t)b * S_ * E_;
#pragma unroll
    for (int k = 0; k < 5; ++k) {
        const int row = wave + k * WAVES;           // waves 0..3 take a 5th row
        if (row < ROWS) {
            const int s_row = s0 - 2 + row;
            const uint32_t lds_off =
                (uint32_t)(uintptr_t)&tile[row * E_ + lane * 4];
            if (0 <= s_row && s_row < S_) {         // wave-uniform branch
                const uint32_t goff =
                    (uint32_t)(((uint32_t)s_row * E_ + lane * 4) * sizeof(float));
                // GVS mode: 64-bit SGPR base plus 32-bit VGPR byte offset, copied to the LDS address in lds_off
                asm volatile("global_load_async_to_lds_b128 %0, %1, %2"
                             :: "v"(lds_off), "v"(goff), "s"(emb_base)
                             : "memory");
            } else {
                *(v4f*)&tile[row * E_ + lane * 4] = (v4f)0.0f;
            }
        }
    }

    // ---- While the async copies are in flight: fetch seq ids and prefetch the
    // gathered U rows (2560 B each) into cache (global_prefetch_b8).
    int vv[ITER];
#pragma unroll
    for (int it = 0; it < ITER; ++it) {
        const int s = s0 + it * WAVES + wave;
        vv[it] = seq[(size_t)b * S_ + s];
    }
#pragma unroll
    for (int it = 0; it < ITER; ++it) {
        // 20 lanes x 128 B = 2560 B = exactly one U row (R_*E_ floats)
        if (lane < 20)
            __builtin_prefetch(U + (size_t)vv[it] * (R_ * E_) + lane * 32, 0, 0);
    }

    // ---- Drain async copies, then make the tile visible to all waves.
    asm volatile("s_wait_asynccnt 0" ::: "memory");
    __syncthreads();

    // ---- Compute: out[b,s,e] = max_j emb[b, s+j-2, e] * U[seq[b,s], j, e]
    // No bounds checks: halo rows in LDS are zero.
#pragma unroll
    for (int it = 0; it < ITER; ++it) {
        const int s_local = it * WAVES + wave;
        const int s = s0 + s_local;
        const float* urow = U + (size_t)vv[it] * (R_ * E_) + lane * 4;
        const float* erow = &tile[s_local * E_ + lane * 4];

        v4f acc = (*(const v4f*)erow) * (*(const v4f*)urow);   // j = 0
#pragma unroll
        for (int j = 1; j < R_; ++j) {
            const v4f e = *(const v4f*)(erow + j * E_);
            const v4f u = *(const v4f*)(urow + j * E_);
            const v4f c = e * u;
            acc.x = fmaxf(acc.x, c.x);
            acc.y = fmaxf(acc.y, c.y);
            acc.z = fmaxf(acc.z, c.z);
            acc.w = fmaxf(acc.w, c.w);
        }
        // Output is write-once / never re-read: stream it past L2 (TH_STORE_NT)
        // so L2 capacity stays dedicated to the 128 MB U gather table.
        __builtin_nontemporal_store(
            acc, (v4f*)&out[((size_t)b * S_ + s) * E_ + lane * 4]);
    }
}

extern "C" void kernel_launch(void* const* d_in, const int* in_sizes, int n_in,
                              void* d_out, int out_size, void* d_ws, size_t ws_size,
                              hipStream_t stream)
{
    const int*   seq     = (const int*)d_in[0];    // (B,S) indices
    const float* seq_emb = (const float*)d_in[1];  // (B,S,E) f32
    const float* U       = (const float*)d_in[2];  // (V,R,E) f32
    float*       out     = (float*)d_out;          // (B,S,E) f32

    const dim3 grid(B_ * (S_ / TS));               // 1024 blocks
    region_embed_kernel<<<grid, THREADS, 0, stream>>>(seq, seq_emb, U, out);
}